// SlidingPosBiases2D_17712445128958
// MI455X (gfx1250) — compile-verified
//
#include <hip/hip_runtime.h>
#include <hip/hip_bf16.h>
#include <cmath>

// SlidingPosBiases2D: out[i,j,k,l] = biases[k-i+R, l-j+R] when in range else 0.
// Pure store-bandwidth problem (340 MB out, ~900 B in). Strategy:
//  - stage 15x15 bias table into LDS via gfx1250 async global->LDS copy
//  - one workgroup per (i,j): writes one contiguous 96x96 tile
//  - 16B non-temporal stores (output >> L2, never re-read)

#define RR 7
#define DD (2 * RR + 1)  // 15

typedef float v4f __attribute__((ext_vector_type(4)));
typedef __attribute__((address_space(1))) int gbl_i32;  // global
typedef __attribute__((address_space(3))) int lds_i32;  // LDS

// Hot kernel, specialized for H == W == 96 (the reference shape).
// Defined first (non-template) so it leads the device assembly.
__global__ __launch_bounds__(256) void sliding_bias_tile96_kernel(
    const float* __restrict__ biases, float* __restrict__ out) {
  constexpr int H = 96, W = 96;
  __shared__ float sb[DD * DD];  // 225 floats = 900 B

  const int t = threadIdx.x;

  // ---- Stage bias table into LDS (CDNA5 async global->LDS path) ----
#if defined(__gfx1250__) && __has_builtin(__builtin_amdgcn_global_load_async_to_lds_b32)
  if (t < DD * DD) {
    float* bsrc = const_cast<float*>(biases) + t;
    __builtin_amdgcn_global_load_async_to_lds_b32(
        (gbl_i32*)bsrc, (lds_i32*)(sb + t), /*imm offset=*/0, /*cpol=*/0);
  }
#if __has_builtin(__builtin_amdgcn_s_wait_asynccnt)
  __builtin_amdgcn_s_wait_asynccnt(0);
#else
  asm volatile("s_wait_asynccnt 0" ::: "memory");
#endif
#else
  if (t < DD * DD) sb[t] = biases[t];
#endif
  __syncthreads();

  // ---- One (i,j) pair per block; tile of H*W contiguous floats ----
  const int bid = blockIdx.x;     // 0 .. H*W-1
  const int i = bid / W;          // compile-time divisor -> mul/shift
  const int j = bid - i * W;
  float* __restrict__ outp = out + (size_t)bid * (size_t)(H * W);

  const int rhbase = RR - i;      // rel_h = k + rhbase
  const int rwbase = RR - j;      // rel_w = l + rwbase

  constexpr int W4 = W / 4;                 // 24 float4 per row
  constexpr int ITERS = (H * W4) / 256;     // 9 iterations, exact

#pragma unroll
  for (int it = 0; it < ITERS; ++it) {
    const int m4 = t + it * 256;            // float4 slot in tile
    const int k = m4 / W4;                  // const divisor
    const int l0 = (m4 - k * W4) * 4;

    v4f v = (v4f)0.0f;
    const int rh = k + rhbase;
    if ((unsigned)rh < (unsigned)DD) {
      const float* __restrict__ row = &sb[rh * DD];
      const int rw = l0 + rwbase;
      if ((unsigned)(rw + 0) < (unsigned)DD) v.x = row[rw + 0];
      if ((unsigned)(rw + 1) < (unsigned)DD) v.y = row[rw + 1];
      if ((unsigned)(rw + 2) < (unsigned)DD) v.z = row[rw + 2];
      if ((unsigned)(rw + 3) < (unsigned)DD) v.w = row[rw + 3];
    }
    // Streaming 16B store: global_store_b128 with TH_STORE_NT
    __builtin_nontemporal_store(v, (v4f*)outp + m4);
  }
}

// Lean 32-bit fallback for non-96 square shapes (reference always uses 96).
__global__ void sliding_bias_generic_kernel(const float* __restrict__ biases,
                                            float* __restrict__ out,
                                            unsigned H, unsigned W, unsigned n) {
  unsigned idx = blockIdx.x * blockDim.x + threadIdx.x;
  if (idx >= n) return;
  unsigned l = idx % W;
  unsigned r = idx / W;
  unsigned k = r % H;
  r /= H;
  unsigned j = r % W;
  unsigned i = r / W;
  int rh = (int)k - (int)i + RR;
  int rw = (int)l - (int)j + RR;
  float v = 0.0f;
  if ((unsigned)rh < (unsigned)DD && (unsigned)rw < (unsigned)DD)
    v = biases[rh * DD + rw];
  __builtin_nontemporal_store(v, out + idx);
}

extern "C" void kernel_launch(void* const* d_in, const int* in_sizes, int n_in,
                              void* d_out, int out_size, void* d_ws, size_t ws_size,
                              hipStream_t stream) {
  const float* biases = (const float*)d_in[0];
  float* out = (float*)d_out;

  // out_size = H*W*H*W with H == W in the reference; recover H on host.
  long long hw = (long long)llround(sqrt((double)out_size));
  int H = (int)llround(sqrt((double)hw));

  if ((long long)H * H == hw && hw * hw == (long long)out_size && H == 96) {
    // 96*96 = 9216 blocks, each writes one contiguous 36 KB tile.
    sliding_bias_tile96_kernel<<<dim3(96 * 96), dim3(256), 0, stream>>>(biases, out);
  } else {
    unsigned W = (unsigned)H;  // reference always square
    unsigned n = (unsigned)out_size;
    unsigned blocks = (n + 255u) / 256u;
    sliding_bias_generic_kernel<<<dim3(blocks), dim3(256), 0, stream>>>(
        biases, out, (unsigned)H, W, n);
  }
}